// NodeApplyModule_44702019616958
// MI455X (gfx1250) — compile-verified
//
#include <hip/hip_runtime.h>
#include <math.h>

#define IN_F 128
#define OUT_F 64
#define NEG_SLOPE 0.01f

typedef __attribute__((ext_vector_type(2))) float v2f;
typedef __attribute__((ext_vector_type(8))) float v8f;

// ---------------------------------------------------------------------------
// Kernel 1: z = h @ W^T via V_WMMA_F32_16X16X4_F32.
// Block = 128 threads = 4 waves. Block computes 16 rows x 64 cols of z.
// Wave w computes the 16x16 tile at columns [16w, 16w+16).
// ---------------------------------------------------------------------------
__global__ __launch_bounds__(128) void gemm_wmma_kernel(
    const float* __restrict__ h, const float* __restrict__ W,
    float* __restrict__ z)
{
    __shared__ float hTile[16 * IN_F];  // 8 KB

    const int nbase = blockIdx.x * 16;

    // Cooperative, coalesced load of 16 h-rows (2048 floats) into LDS.
    const float4* hsrc = (const float4*)(h + (size_t)nbase * IN_F);
    float4* hdst = (float4*)hTile;
    for (int i = threadIdx.x; i < (16 * IN_F) / 4; i += 128)
        hdst[i] = hsrc[i];
    __syncthreads();

    const int wave    = threadIdx.x >> 5;
    const int lane    = threadIdx.x & 31;
    const int halfSel = lane >> 4;     // 0: lanes 0-15, 1: lanes 16-31
    const int l15     = lane & 15;
    const int klo     = halfSel * 2;   // A/B fragment K sub-offset
    const int obase   = wave * 16;

    // A fragment row (M = l15) comes from LDS; B fragment column (N = l15)
    // is row (obase + l15) of W since B[k][o] = W[o][k].
    const float* arow = hTile + l15 * IN_F;
    const float* wrow = W + (size_t)(obase + l15) * IN_F;

    v8f acc = {};
    #pragma unroll 4
    for (int kb = 0; kb < IN_F; kb += 4) {
        v2f a, b;
        a.x = arow[kb + klo];
        a.y = arow[kb + klo + 1];
        b.x = wrow[kb + klo];
        b.y = wrow[kb + klo + 1];
        acc = __builtin_amdgcn_wmma_f32_16x16x4_f32(
            /*neg_a=*/false, a, /*neg_b=*/false, b,
            /*c_mod=*/(short)0, acc, /*reuse_a=*/false, /*reuse_b=*/false);
    }

    // D layout: VGPR r -> M = r (lanes 0-15) or r+8 (lanes 16-31), N = l15.
    const int row0 = nbase + halfSel * 8;
    const int col  = obase + l15;
    #pragma unroll
    for (int r = 0; r < 8; ++r)
        z[(size_t)(row0 + r) * OUT_F + col] = acc[r];
}

// ---------------------------------------------------------------------------
// Kernel 2: per-node attention scores s_l = z.a_l, s_r = z.a_r; init m, denom.
// ---------------------------------------------------------------------------
__global__ void node_prep_kernel(
    const float* __restrict__ z, const float* __restrict__ W_attn,
    float* __restrict__ s_l, float* __restrict__ s_r,
    float* __restrict__ m, float* __restrict__ denom, int n_nodes)
{
    int n = blockIdx.x * blockDim.x + threadIdx.x;
    if (n >= n_nodes) return;
    const float* zr = z + (size_t)n * OUT_F;
    float al = 0.f, ar = 0.f;
    #pragma unroll 8
    for (int o = 0; o < OUT_F; ++o) {
        float zv = zr[o];
        al += zv * W_attn[o];          // uniform -> scalar loads
        ar += zv * W_attn[OUT_F + o];
    }
    s_l[n] = al;
    s_r[n] = ar;
    m[n] = -INFINITY;
    denom[n] = 0.f;
}

// ---------------------------------------------------------------------------
// Kernel 3: zero the output accumulator.
// ---------------------------------------------------------------------------
__global__ void zero_kernel(float4* __restrict__ out4, int n4)
{
    int i = blockIdx.x * blockDim.x + threadIdx.x;
    if (i < n4) {
        float4 zf = {0.f, 0.f, 0.f, 0.f};
        out4[i] = zf;
    }
}

// Exact float atomic-max via ordered-integer mapping.
__device__ inline void atomicMaxFloat(float* addr, float val)
{
    if (val >= 0.0f)
        atomicMax((int*)addr, __float_as_int(val));
    else
        atomicMin((unsigned int*)addr, __float_as_uint(val));
}

// ---------------------------------------------------------------------------
// Kernel 4: edge logits + segment max over dst.
// ---------------------------------------------------------------------------
__global__ void edge_logits_kernel(
    const int* __restrict__ src, const int* __restrict__ dst,
    const float* __restrict__ s_l, const float* __restrict__ s_r,
    float* __restrict__ e_buf, float* __restrict__ m, int n_edges)
{
    int k = blockIdx.x * blockDim.x + threadIdx.x;
    if (k >= n_edges) return;
    float v = s_l[src[k]] + s_r[dst[k]];
    float e = (v > 0.f) ? v : NEG_SLOPE * v;   // leaky_relu
    e_buf[k] = e;
    atomicMaxFloat(&m[dst[k]], e);
}

// ---------------------------------------------------------------------------
// Kernel 5: exp(e - m[dst]) and segment sum of denominators.
// ---------------------------------------------------------------------------
__global__ void edge_exp_kernel(
    const int* __restrict__ dst, float* __restrict__ e_buf,
    const float* __restrict__ m, float* __restrict__ denom, int n_edges)
{
    int k = blockIdx.x * blockDim.x + threadIdx.x;
    if (k >= n_edges) return;
    int d = dst[k];
    float ex = __expf(e_buf[k] - m[d]);
    e_buf[k] = ex;
    atomicAdd(&denom[d], ex);
}

// ---------------------------------------------------------------------------
// Kernel 6: out[dst] += alpha * z[src]. 16 lanes per edge, 4 floats per lane.
// z and out are L2-resident (12.8 MB each << 192 MB L2).
// ---------------------------------------------------------------------------
__global__ void edge_scatter_kernel(
    const int* __restrict__ src, const int* __restrict__ dst,
    const float* __restrict__ e_buf, const float* __restrict__ denom,
    const float* __restrict__ z, float* __restrict__ out, int n_edges)
{
    int tid = blockIdx.x * blockDim.x + threadIdx.x;
    int k = tid >> 4;
    if (k >= n_edges) return;
    int c0 = (tid & 15) * 4;
    int s = src[k], d = dst[k];
    float alpha = e_buf[k] / denom[d];
    float4 zv = *(const float4*)(z + (size_t)s * OUT_F + c0);
    float* o = out + (size_t)d * OUT_F + c0;
    atomicAdd(o + 0, alpha * zv.x);
    atomicAdd(o + 1, alpha * zv.y);
    atomicAdd(o + 2, alpha * zv.z);
    atomicAdd(o + 3, alpha * zv.w);
}

// ---------------------------------------------------------------------------
extern "C" void kernel_launch(void* const* d_in, const int* in_sizes, int n_in,
                              void* d_out, int out_size, void* d_ws, size_t ws_size,
                              hipStream_t stream)
{
    const float* h      = (const float*)d_in[0];   // N x 128
    const float* W_fc   = (const float*)d_in[1];   // 64 x 128
    const float* W_attn = (const float*)d_in[2];   // 128
    const int*   src    = (const int*)d_in[3];     // E
    const int*   dst    = (const int*)d_in[4];     // E

    const int N = in_sizes[0] / IN_F;              // 50000 (multiple of 16)
    const int E = in_sizes[3];                     // 800000
    float* out = (float*)d_out;                    // N x 64

    // Workspace layout (256B aligned regions).
    auto align256 = [](size_t x) { return (x + 255) & ~(size_t)255; };
    char* ws = (char*)d_ws;
    size_t off = 0;
    float* z     = (float*)(ws + off); off += align256((size_t)N * OUT_F * 4);
    float* s_l   = (float*)(ws + off); off += align256((size_t)N * 4);
    float* s_r   = (float*)(ws + off); off += align256((size_t)N * 4);
    float* m     = (float*)(ws + off); off += align256((size_t)N * 4);
    float* denom = (float*)(ws + off); off += align256((size_t)N * 4);
    float* e_buf = (float*)(ws + off); off += align256((size_t)E * 4);
    (void)ws_size; (void)n_in;

    // 1) z = h @ W^T (WMMA f32)
    gemm_wmma_kernel<<<N / 16, 128, 0, stream>>>(h, W_fc, z);

    // 2) per-node scores + init
    node_prep_kernel<<<(N + 255) / 256, 256, 0, stream>>>(
        z, W_attn, s_l, s_r, m, denom, N);

    // 3) zero output
    int n4 = out_size / 4;
    zero_kernel<<<(n4 + 255) / 256, 256, 0, stream>>>((float4*)out, n4);

    // 4) edge logits + segment max
    edge_logits_kernel<<<(E + 255) / 256, 256, 0, stream>>>(
        src, dst, s_l, s_r, e_buf, m, E);

    // 5) exp + denominator sum
    edge_exp_kernel<<<(E + 255) / 256, 256, 0, stream>>>(
        dst, e_buf, m, denom, E);

    // 6) weighted scatter
    long long total = (long long)E * 16;
    edge_scatter_kernel<<<(int)((total + 255) / 256), 256, 0, stream>>>(
        src, dst, e_buf, denom, z, out, E);
}